// NodeNet_46273977647808
// MI455X (gfx1250) — compile-verified
//
#include <hip/hip_runtime.h>
#include <math.h>

typedef float v2f __attribute__((ext_vector_type(2)));
typedef float v8f __attribute__((ext_vector_type(8)));

#define N_ROWS 8192
#define E_DIM  16384
#define D_DIM  256
#define K1_CHUNKS 8   // n split: 8 chunks of 1024
#define K2_CHUNKS 8   // j split: 8 chunks of 2048

// ---------------------------------------------------------------------------
// K1: partial bo8[j,k] = sum_{n in chunk} Ro[n,j] * X[n,k]  (k<8; cols 8..15 of
// the WMMA tile are dead — lanes 8..15 load replicated data and the result is
// discarded at store time, so no masking/branching in the hot loop).
// A = Ro^T tile (16 j x 4 n), B = X tile (4 n x 16 k), D accum 16x16 f32.
// ---------------------------------------------------------------------------
__global__ __launch_bounds__(256) void k1_bo_partial(const float* __restrict__ Ro,
                                                     const float* __restrict__ X,
                                                     float* __restrict__ pb) {
  const int tid  = blockIdx.x * blockDim.x + threadIdx.x;
  const int w    = tid >> 5;
  const int lane = tid & 31;
  const int lo   = lane & 15;
  const int hi   = lane >> 4;
  const int kb   = 2 * hi;

  const int jw    = w & (E_DIM / 16 - 1);
  const int chunk = w / (E_DIM / 16);
  const int j0    = jw * 16;
  const int nbeg  = chunk * (N_ROWS / K1_CHUNKS);

  // A[M=lo][K=kb+t] = Ro[n+kb+t][j0+lo] ; B[K=kb+t][N] = X[n+kb+t][N&7]
  const float* pA = Ro + (size_t)(nbeg + kb) * E_DIM + j0 + lo;
  const float* pB = X  + (size_t)(nbeg + kb) * D_DIM + (lo & 7);

  v8f c = {0.f, 0.f, 0.f, 0.f, 0.f, 0.f, 0.f, 0.f};

  #pragma unroll 8
  for (int it = 0; it < (N_ROWS / K1_CHUNKS) / 4; ++it) {
    v2f a, b;
    a.x = pA[0];
    a.y = pA[E_DIM];
    b.x = pB[0];
    b.y = pB[D_DIM];
    c = __builtin_amdgcn_wmma_f32_16x16x4_f32(false, a, false, b, (short)0, c,
                                              false, false);
    pA += 4 * E_DIM;
    pB += 4 * D_DIM;
  }

  // D layout: VGPR r -> (M=r, N=lane) lanes 0-15 ; (M=8+r, N=lane-16) lanes 16-31
  float* outp = pb + (size_t)chunk * (E_DIM * 8);
  if (lo < 8) {
    #pragma unroll
    for (int r = 0; r < 8; ++r)
      outp[(size_t)(j0 + r + 8 * hi) * 8 + lo] = c[r];
  }
}

// boe[j,k] = e[j] * sum_chunk pb[chunk][j,k]
__global__ __launch_bounds__(256) void k1_reduce(const float* __restrict__ pb,
                                                 const float* __restrict__ e,
                                                 float* __restrict__ boe) {
  const int i = blockIdx.x * blockDim.x + threadIdx.x;  // 0 .. E*8-1
  float s = 0.f;
  #pragma unroll
  for (int cgi = 0; cgi < K1_CHUNKS; ++cgi) s += pb[(size_t)cgi * (E_DIM * 8) + i];
  boe[i] = s * e[i >> 3];
}

// ---------------------------------------------------------------------------
// K2: partial mi8[n,k] = sum_{j in chunk} Ri[n,j] * boe[j,k]
// A = Ri tile (16 n x 4 j), B = boe tile (4 j x 16 k), cols 8..15 dead.
// ---------------------------------------------------------------------------
__global__ __launch_bounds__(256) void k2_mi_partial(const float* __restrict__ Ri,
                                                     const float* __restrict__ boe,
                                                     float* __restrict__ pm) {
  const int tid  = blockIdx.x * blockDim.x + threadIdx.x;
  const int w    = tid >> 5;
  const int lane = tid & 31;
  const int lo   = lane & 15;
  const int hi   = lane >> 4;
  const int kb   = 2 * hi;

  const int nw    = w & (N_ROWS / 16 - 1);
  const int chunk = w / (N_ROWS / 16);
  const int n0    = nw * 16;
  const int jbeg  = chunk * (E_DIM / K2_CHUNKS);

  // A[M=lo][K=kb..kb+1] = Ri[n0+lo][j+kb .. j+kb+1]  (one 8B load)
  const float* pA = Ri  + (size_t)(n0 + lo) * E_DIM + jbeg + kb;
  const float* pB = boe + (size_t)(jbeg + kb) * 8 + (lo & 7);

  v8f c = {0.f, 0.f, 0.f, 0.f, 0.f, 0.f, 0.f, 0.f};

  #pragma unroll 8
  for (int it = 0; it < (E_DIM / K2_CHUNKS) / 4; ++it) {
    v2f a = *(const v2f*)pA;
    v2f b;
    b.x = pB[0];
    b.y = pB[8];
    c = __builtin_amdgcn_wmma_f32_16x16x4_f32(false, a, false, b, (short)0, c,
                                              false, false);
    pA += 4;
    pB += 32;
  }

  float* outp = pm + (size_t)chunk * (N_ROWS * 8);
  if (lo < 8) {
    #pragma unroll
    for (int r = 0; r < 8; ++r)
      outp[(size_t)(n0 + r + 8 * hi) * 8 + lo] = c[r];
  }
}

__global__ __launch_bounds__(256) void k2_reduce(const float* __restrict__ pm,
                                                 float* __restrict__ mi8) {
  const int i = blockIdx.x * blockDim.x + threadIdx.x;  // 0 .. N*8-1
  float s = 0.f;
  #pragma unroll
  for (int cgi = 0; cgi < K2_CHUNKS; ++cgi) s += pm[(size_t)cgi * (N_ROWS * 8) + i];
  mi8[i] = s;
}

// ---------------------------------------------------------------------------
// K3: 8-qubit TTN circuit, one wave32 per row. 256 amplitudes = 8 regs/lane.
// idx = r*32 + lane; wire w <-> bit p = 7-w. Wires 0,1,2 -> reg bits (m=4,2,1);
// wires 3..7 -> lane bits p=4..0.
// ---------------------------------------------------------------------------
#define RY_REG(m, t) do {                                                   \
    float c_ = cosf((t) * 0.5f), s_ = sinf((t) * 0.5f);                     \
    _Pragma("unroll")                                                       \
    for (int r_ = 0; r_ < 8; ++r_) if (!(r_ & (m))) {                       \
      float a0_ = amp[r_], a1_ = amp[r_ | (m)];                             \
      amp[r_]       = c_ * a0_ - s_ * a1_;                                  \
      amp[r_ | (m)] = s_ * a0_ + c_ * a1_;                                  \
    }                                                                       \
  } while (0)

#define RY_LANE(p, t) do {                                                  \
    float c_ = cosf((t) * 0.5f), s_ = sinf((t) * 0.5f);                     \
    float ss_ = ((lane >> (p)) & 1) ? s_ : -s_;                             \
    _Pragma("unroll")                                                       \
    for (int r_ = 0; r_ < 8; ++r_) {                                        \
      float v_  = amp[r_];                                                  \
      float pv_ = __shfl_xor(v_, 1 << (p), 32);                             \
      amp[r_] = c_ * v_ + ss_ * pv_;                                        \
    }                                                                       \
  } while (0)

#define CNOT_RR(mc, mt) do {                                                \
    _Pragma("unroll")                                                       \
    for (int r_ = 0; r_ < 8; ++r_) if ((r_ & (mc)) && !(r_ & (mt))) {       \
      float a0_ = amp[r_]; amp[r_] = amp[r_ | (mt)]; amp[r_ | (mt)] = a0_;  \
    }                                                                       \
  } while (0)

#define CNOT_LL(pc, pt) do {                                                \
    bool ctl_ = (lane >> (pc)) & 1;                                         \
    _Pragma("unroll")                                                       \
    for (int r_ = 0; r_ < 8; ++r_) {                                        \
      float v_  = amp[r_];                                                  \
      float pv_ = __shfl_xor(v_, 1 << (pt), 32);                            \
      amp[r_] = ctl_ ? pv_ : v_;                                            \
    }                                                                       \
  } while (0)

#define CNOT_LC_RT(pc, mt) do {                                             \
    bool ctl_ = (lane >> (pc)) & 1;                                         \
    _Pragma("unroll")                                                       \
    for (int r_ = 0; r_ < 8; ++r_) if (!(r_ & (mt))) {                      \
      float a0_ = amp[r_], a1_ = amp[r_ | (mt)];                            \
      amp[r_]        = ctl_ ? a1_ : a0_;                                    \
      amp[r_ | (mt)] = ctl_ ? a0_ : a1_;                                    \
    }                                                                       \
  } while (0)

#define CNOT_RC_LT(mc, pt) do {                                             \
    _Pragma("unroll")                                                       \
    for (int r_ = 0; r_ < 8; ++r_) if (r_ & (mc)) {                         \
      amp[r_] = __shfl_xor(amp[r_], 1 << (pt), 32);                         \
    }                                                                       \
  } while (0)

__global__ __launch_bounds__(256) void k3_circuit(const float* __restrict__ mi8,
                                                  const float* __restrict__ theta,
                                                  float* __restrict__ out) {
  const int tid  = blockIdx.x * blockDim.x + threadIdx.x;
  const int w    = tid >> 5;   // row index
  const int lane = tid & 31;

  const float* ang = mi8 + (size_t)w * 8;
  float th[15];
  #pragma unroll
  for (int i = 0; i < 15; ++i) th[i] = theta[i];

  float amp[8];
  #pragma unroll
  for (int r = 1; r < 8; ++r) amp[r] = 0.f;
  amp[0] = (lane == 0) ? 1.f : 0.f;

  // initial per-wire RY from angles
  RY_REG(4, ang[0]);   // wire 0
  RY_REG(2, ang[1]);   // wire 1
  RY_REG(1, ang[2]);   // wire 2
  RY_LANE(4, ang[3]);  // wire 3
  RY_LANE(3, ang[4]);  // wire 4
  RY_LANE(2, ang[5]);  // wire 5
  RY_LANE(1, ang[6]);  // wire 6
  RY_LANE(0, ang[7]);  // wire 7

  RY_REG(4, th[0]);  RY_REG(2, th[1]);   CNOT_RR(4, 2);     // CNOT(0,1)
  RY_REG(1, th[2]);  RY_LANE(4, th[3]);  CNOT_LC_RT(4, 1);  // CNOT(3,2)
  RY_LANE(3, th[4]); RY_LANE(2, th[5]);  CNOT_LL(3, 2);     // CNOT(4,5)
  RY_LANE(1, th[6]); RY_LANE(0, th[7]);  CNOT_LL(0, 1);     // CNOT(7,6)
  RY_REG(2, th[8]);  RY_REG(1, th[9]);   CNOT_RR(2, 1);     // CNOT(1,2)
  RY_LANE(2, th[10]); RY_LANE(1, th[11]); CNOT_LL(1, 2);    // CNOT(6,5)
  RY_REG(1, th[12]); RY_LANE(2, th[13]); CNOT_RC_LT(1, 2);  // CNOT(2,5)
  RY_LANE(2, th[14]);                                       // RY wire 5

  // z = sum |amp|^2 * (+1 if wire5 bit (lane bit 2) == 0 else -1)
  float loc = 0.f;
  #pragma unroll
  for (int r = 0; r < 8; ++r) loc += amp[r] * amp[r];
  loc = (lane & 4) ? -loc : loc;
  #pragma unroll
  for (int off = 16; off >= 1; off >>= 1) loc += __shfl_xor(loc, off, 32);

  if (lane == 0) out[w] = (1.0f - loc) * 0.5f;
}

// ---------------------------------------------------------------------------
extern "C" void kernel_launch(void* const* d_in, const int* in_sizes, int n_in,
                              void* d_out, int out_size, void* d_ws, size_t ws_size,
                              hipStream_t stream) {
  const float* X     = (const float*)d_in[0];  // (8192, 256)
  const float* e     = (const float*)d_in[1];  // (16384,)
  const float* Ri    = (const float*)d_in[2];  // (8192, 16384)
  const float* Ro    = (const float*)d_in[3];  // (8192, 16384)
  const float* theta = (const float*)d_in[4];  // (15,)
  float* out = (float*)d_out;                  // (8192,)

  char* ws = (char*)d_ws;
  size_t off = 0;
  float* pb  = (float*)(ws + off); off += (size_t)K1_CHUNKS * E_DIM * 8 * sizeof(float); // 4 MB
  float* boe = (float*)(ws + off); off += (size_t)E_DIM * 8 * sizeof(float);             // 512 KB
  float* pm  = (float*)(ws + off); off += (size_t)K2_CHUNKS * N_ROWS * 8 * sizeof(float);// 2 MB
  float* mi8 = (float*)(ws + off);                                                       // 256 KB
  (void)in_sizes; (void)n_in; (void)out_size; (void)ws_size;

  // K1: 8192 waves = 262144 threads
  k1_bo_partial<<<(K1_CHUNKS * (E_DIM / 16) * 32) / 256, 256, 0, stream>>>(Ro, X, pb);
  k1_reduce<<<(E_DIM * 8) / 256, 256, 0, stream>>>(pb, e, boe);
  // K2: 4096 waves = 131072 threads
  k2_mi_partial<<<(K2_CHUNKS * (N_ROWS / 16) * 32) / 256, 256, 0, stream>>>(Ri, boe, pm);
  k2_reduce<<<(N_ROWS * 8) / 256, 256, 0, stream>>>(pm, mi8);
  // K3: one wave per row
  k3_circuit<<<(N_ROWS * 32) / 256, 256, 0, stream>>>(mi8, theta, out);
}